// MNTMHeadOp_59983513256195
// MI455X (gfx1250) — compile-verified
//
#include <hip/hip_runtime.h>
#include <hip/hip_bf16.h>
#include <math.h>

// ---------------------------------------------------------------------------
// NTM head op for MI455X (gfx1250, wave32).
// Heavy phase = per-batch GEMM [8x1024] x [1024x4096] in fp32 via
// V_WMMA_F32_16X16X4_F32, fused with the ||Mem||^2 reduction so the 512 MiB
// Prev_Memory tensor is streamed from HBM exactly once (HBM-bound, ~23 us).
// ---------------------------------------------------------------------------

typedef float v2f __attribute__((ext_vector_type(2)));
typedef float v8f __attribute__((ext_vector_type(8)));

#define EPS_F 1e-8f
#define B_SZ 32
#define N_SZ 4096
#define H_SZ 8
#define KDIM 1024   // M1*M2 = 32*32

__device__ __forceinline__ float softplus20(float x) {
    return (x > 20.0f) ? x : log1pf(expf(fminf(x, 20.0f)));
}
__device__ __forceinline__ float clamp080(float x) {
    return fminf(fmaxf(x, 0.0f), 80.0f);
}
__device__ __forceinline__ float sigmoidf_(float x) {
    return 1.0f / (1.0f + expf(-x));
}

// ---------------------------------------------------------------------------
// K0: zero accumulators; per-(b,h) params: beta, g, gamma, shift-softmax.
// 256 threads, one block. Param arrays indexed [b*8 + h].
// ---------------------------------------------------------------------------
__global__ __launch_bounds__(256) void k0_init(
    const float* __restrict__ s_t,        // [B,3,H]
    const float* __restrict__ bgg,        // [B,3,H]
    float* __restrict__ normKsq, float* __restrict__ sumEx, float* __restrict__ sumSh,
    float* __restrict__ betaA, float* __restrict__ gA, float* __restrict__ gammaA,
    float* __restrict__ ssm)              // [B*8][3]
{
    const int tid = threadIdx.x;          // tid = b*8 + h
    normKsq[tid] = 0.0f; sumEx[tid] = 0.0f; sumSh[tid] = 0.0f;
    const int b = tid >> 3, h = tid & 7;
    const float x0 = bgg[b * 24 + 0 * 8 + h];
    const float x1 = bgg[b * 24 + 1 * 8 + h];
    const float x2 = bgg[b * 24 + 2 * 8 + h];
    betaA[tid]  = 1.0f + softplus20(x0);
    gA[tid]     = sigmoidf_(x1);
    gammaA[tid] = 1.0f + softplus20(x2);
    float e0 = expf(clamp080(s_t[b * 24 + 0 * 8 + h]));
    float e1 = expf(clamp080(s_t[b * 24 + 1 * 8 + h]));
    float e2 = expf(clamp080(s_t[b * 24 + 2 * 8 + h]));
    float inv = 1.0f / (e0 + e1 + e2 + EPS_F);
    ssm[tid * 3 + 0] = e0 * inv;
    ssm[tid * 3 + 1] = e1 * inv;
    ssm[tid * 3 + 2] = e2 * inv;
}

// ---------------------------------------------------------------------------
// K1: (a) Kh = tanh(K_t) into zero-padded [B][16][1024] layout + ||Kh||^2
//     (b) E/A matrices (sigmoid/tanh) written directly to d_out.
// Threads: 16384 for (a), 262144 for (b)  -> 1088 blocks x 256.
// ---------------------------------------------------------------------------
__global__ __launch_bounds__(256) void k1_pre(
    const float* __restrict__ K_t,        // [B,32,H*32]
    const float* __restrict__ AE,         // [B,32,8*32]
    float* __restrict__ Khpad,            // [B,16,1024]
    float* __restrict__ normKsq,          // [B*8]
    float* __restrict__ Eout,             // [4,B,32,32]
    float* __restrict__ Aout)             // [4,B,32,32]
{
    const int t = blockIdx.x * 256 + threadIdx.x;
    if (t < 16384) {                      // one thread per (b, h_pad, i) row
        const int b = t >> 9;             // /(16*32)
        const int h = (t >> 5) & 15;
        const int i = t & 31;
        float* dst = Khpad + ((size_t)(b * 16 + h)) * KDIM + i * 32;
        if (h < 8) {
            const float* src = K_t + ((size_t)(b * 32 + i)) * 256 + h * 32;
            float acc = 0.0f;
            #pragma unroll
            for (int j = 0; j < 32; ++j) {
                float v = tanhf(src[j]);
                dst[j] = v;
                acc = fmaf(v, v, acc);
            }
            atomicAdd(&normKsq[b * 8 + h], acc);
        } else {                          // zero padding rows (heads 8..15)
            #pragma unroll
            for (int j = 0; j < 32; ++j) dst[j] = 0.0f;
        }
    } else {
        const int u = t - 16384;
        if (u < 262144) {                 // one thread per A/E element
            const float v = AE[u];
            const int j = u & 31;
            const int w = (u >> 5) & 7;
            const int i = (u >> 8) & 31;
            const int b = u >> 13;
            if (w < 4) {                  // A_mats = tanh
                Aout[(((w * B_SZ + b) * 32 + i) << 5) + j] = tanhf(v);
            } else {                      // E_mats = sigmoid
                Eout[((((w - 4) * B_SZ + b) * 32 + i) << 5) + j] = sigmoidf_(v);
            }
        }
    }
}

// ---------------------------------------------------------------------------
// K2: WMMA pass. One wave per (b, 16-slot tile): 8192 waves.
//   D[16x16] = Kh[16heads x K] * Mem[K x 16slots], K accumulated 4 at a time
//   via V_WMMA_F32_16X16X4_F32. ||Mem||^2 fused from the same registers.
//   Epilogue: sim -> exp(clip(beta*sim,0,80)) -> exBuf + per-(h,b) sums.
//
// A operand (32-bit 16x4): lane l holds head (l&15), {k0+2*(l>>4), +1}.
// B operand (symmetric):   lane l holds slot (l&15), same k offsets.
// D:  lanes 0..15, vgpr r = num[head r][slot lane]; lanes 16..31 = pad heads.
// ---------------------------------------------------------------------------
__global__ __launch_bounds__(256) void k2_wmma(
    const float* __restrict__ Mem,        // [B,N,32,32]
    const float* __restrict__ Khpad,      // [B,16,1024]
    const float* __restrict__ normKsq,    // [B*8]
    const float* __restrict__ betaA,      // [B*8]
    float* __restrict__ exBuf,            // [H,B,N]
    float* __restrict__ sumEx)            // [B*8]
{
    const int wave = blockIdx.x * (blockDim.x >> 5) + (threadIdx.x >> 5);
    const int lane = threadIdx.x & 31;
    const int b    = wave >> 8;           // 256 tiles per batch
    const int tile = wave & 255;
    const int n0   = tile << 4;
    const int lm   = lane & 15;           // head for A, slot for B, slot for D
    const int hi   = lane >> 4;           // selects K sub-pair {0,1} vs {2,3}

    const float* memRow = Mem + (((size_t)b << 12) + (size_t)(n0 + lm)) * KDIM + 2 * hi;
    const float* khRow  = Khpad + ((size_t)(b * 16 + lm)) * KDIM + 2 * hi;

    v8f c = {0.f, 0.f, 0.f, 0.f, 0.f, 0.f, 0.f, 0.f};
    float nsq = 0.0f;

    #pragma unroll 8
    for (int k = 0; k < KDIM; k += 4) {
        v2f a  = *(const v2f*)(khRow + k);
        v2f bm = *(const v2f*)(memRow + k);
        nsq = fmaf(bm.x, bm.x, nsq);
        nsq = fmaf(bm.y, bm.y, nsq);
        // D = A x B + C   (f32 16x16x4)
        c = __builtin_amdgcn_wmma_f32_16x16x4_f32(
                /*neg_a=*/false, a, /*neg_b=*/false, bm,
                /*c_mod=*/(short)0, c, /*reuse_a=*/false, /*reuse_b=*/false);
    }

    // lanes l and l^16 each hold half of slot (l&15)'s sum of squares
    const float normMem = sqrtf(nsq + __shfl_xor(nsq, 16));

    #pragma unroll
    for (int r = 0; r < 8; ++r) {         // r = head; valid on lanes 0..15
        const float nk   = sqrtf(normKsq[b * 8 + r]);
        const float beta = betaA[b * 8 + r];
        const float sim  = c[r] / (normMem * nk + EPS_F);
        const float e    = expf(clamp080(beta * sim));
        if (lane < 16)
            exBuf[(((size_t)r * B_SZ + b) << 12) + (size_t)(n0 + lm)] = e;
        // 16-lane tree reduce (xor masks keep low/high halves separate)
        float s = (lane < 16) ? e : 0.0f;
        s += __shfl_xor(s, 1);
        s += __shfl_xor(s, 2);
        s += __shfl_xor(s, 4);
        s += __shfl_xor(s, 8);
        if (lane == 0) atomicAdd(&sumEx[b * 8 + r], s);
    }
}

// ---------------------------------------------------------------------------
// K3: W_g = g*W_c + (1-g)*Prev_W recomputed at n-1,n,n+1 from exBuf;
//     3-tap circular shift; W_sh = pow(W_hat, gamma) -> d_out (W region);
//     block reduce + atomicAdd into sumSh. Each block covers one (h,b).
// ---------------------------------------------------------------------------
__global__ __launch_bounds__(256) void k3_shift(
    const float* __restrict__ exBuf,      // [H,B,N]
    const float* __restrict__ PrevW,      // [H,B,N]
    const float* __restrict__ sumEx, const float* __restrict__ gA,
    const float* __restrict__ gammaA, const float* __restrict__ ssm,
    float* __restrict__ Wout,             // [H,B,N] (holds W_sh after K3)
    float* __restrict__ sumSh)
{
    const int idx = blockIdx.x * 256 + threadIdx.x;
    const int n  = idx & (N_SZ - 1);
    const int hb = idx >> 12;             // h*32 + b
    const int h = hb >> 5, b = hb & 31;
    const size_t base = (size_t)hb << 12;
    const int p = b * 8 + h;

    const float g   = gA[p];
    const float inv = g / (sumEx[p] + EPS_F);
    const float om  = 1.0f - g;
    const int nm = (n + N_SZ - 1) & (N_SZ - 1);
    const int np = (n + 1) & (N_SZ - 1);

    const float w_m = inv * exBuf[base + nm] + om * PrevW[base + nm];
    const float w_0 = inv * exBuf[base + n ] + om * PrevW[base + n ];
    const float w_p = inv * exBuf[base + np] + om * PrevW[base + np];

    const float* s = ssm + p * 3;
    const float what = s[0] * w_m + s[1] * w_0 + s[2] * w_p;   // roll semantics
    const float wsh  = powf(what, gammaA[p]);
    Wout[base + n] = wsh;

    __shared__ float red[256];
    red[threadIdx.x] = wsh;
    __syncthreads();
    #pragma unroll
    for (int st = 128; st > 0; st >>= 1) {
        if (threadIdx.x < st) red[threadIdx.x] += red[threadIdx.x + st];
        __syncthreads();
    }
    if (threadIdx.x == 0) atomicAdd(&sumSh[p], red[0]);
}

// ---------------------------------------------------------------------------
// K4: in-place normalize W = W_sh / (sum + eps).
// ---------------------------------------------------------------------------
__global__ __launch_bounds__(256) void k4_norm(
    float* __restrict__ W, const float* __restrict__ sumSh)
{
    const int idx = blockIdx.x * 256 + threadIdx.x;
    const int hb = idx >> 12;
    const int h = hb >> 5, b = hb & 31;
    W[idx] = W[idx] / (sumSh[b * 8 + h] + EPS_F);
}

// ---------------------------------------------------------------------------
// Launch.  d_in order: K_t, A_t_and_E_t, s_t, beta_g_gamma, Prev_Memory, Prev_W
// d_out: W[8*32*4096] | E_mats[4*32*32*32] | A_mats[4*32*32*32]
// ---------------------------------------------------------------------------
extern "C" void kernel_launch(void* const* d_in, const int* in_sizes, int n_in,
                              void* d_out, int out_size, void* d_ws, size_t ws_size,
                              hipStream_t stream) {
    const float* K_t  = (const float*)d_in[0];
    const float* AE   = (const float*)d_in[1];
    const float* s_t  = (const float*)d_in[2];
    const float* bgg  = (const float*)d_in[3];
    const float* Mem  = (const float*)d_in[4];
    const float* PrW  = (const float*)d_in[5];

    float* W    = (float*)d_out;                 // 1,048,576
    float* Eout = W + (size_t)H_SZ * B_SZ * N_SZ;        // +1,048,576
    float* Aout = Eout + 4 * B_SZ * 32 * 32;             // +131,072

    // Workspace layout (floats)
    float* wsf     = (float*)d_ws;
    float* Khpad   = wsf;                         // 32*16*1024 = 524288
    float* normKsq = Khpad + 524288;              // 256
    float* sumEx   = normKsq + 256;               // 256
    float* sumSh   = sumEx + 256;                 // 256
    float* betaA   = sumSh + 256;                 // 256
    float* gA      = betaA + 256;                 // 256
    float* gammaA  = gA + 256;                    // 256
    float* ssm     = gammaA + 256;                // 768
    float* exBuf   = ssm + 768;                   // 8*32*4096 = 1048576

    k0_init<<<1, 256, 0, stream>>>(s_t, bgg, normKsq, sumEx, sumSh,
                                   betaA, gA, gammaA, ssm);
    k1_pre<<<(16384 + 262144) / 256, 256, 0, stream>>>(K_t, AE, Khpad, normKsq,
                                                       Eout, Aout);
    // 8192 tiles, one wave32 each -> 1024 blocks of 8 waves
    k2_wmma<<<1024, 256, 0, stream>>>(Mem, Khpad, normKsq, betaA, exBuf, sumEx);
    k3_shift<<<(H_SZ * B_SZ * N_SZ) / 256, 256, 0, stream>>>(exBuf, PrW, sumEx, gA,
                                                             gammaA, ssm, W, sumSh);
    k4_norm<<<(H_SZ * B_SZ * N_SZ) / 256, 256, 0, stream>>>(W, sumSh);
}